// Uiqi_80874234184238
// MI455X (gfx1250) — compile-verified
//
#include <hip/hip_runtime.h>

// UIQI (universal image quality index) for B=16,C=5,H=W=512, 32x32 circular box filter.
// Pass 1: horizontal 32-tap circular window sums for (img1, img2, img1*img2) via
//         banded-matrix WMMA (v_wmma_f32_16x16x4_f32, exact f32 accumulate).
// Pass 2: vertical 32-tap running sums + pointwise q + weighted deterministic reduce.
// Pass 3: sum 80 per-plane partials -> mean.
// Workspace: 3 * 80 * 512*512 * 4B = 251,658,240 B  + 640 B partials.

#define HH 512
#define WW 512
#define PLANES 80
#define PLANE_ELEMS (HH * WW)
#define EPSQ 1e-8f

typedef __attribute__((ext_vector_type(2))) float v2f;
typedef __attribute__((ext_vector_type(8))) float v8f;

// Banded stacked weight W[k][n], k in [0,48), n in [0,16):
//   k<16  : tile t-1 contribution -> lower triangular (n <= k)
//   k<32  : tile t   contribution -> all ones
//   k>=32 : tile t+1 contribution -> strictly upper (n >= k-31)
__device__ __forceinline__ float wband(int k, int n) {
  if (k < 16) return (n <= k) ? 1.f : 0.f;
  if (k < 32) return 1.f;
  return (n >= k - 31) ? 1.f : 0.f;
}

__global__ __launch_bounds__(128)
void hpass_kernel(const float* __restrict__ img1,
                  const float* __restrict__ img2,
                  float* __restrict__ h1,
                  float* __restrict__ h2,
                  float* __restrict__ h12)
{
  // 16 rows x 96 cols staged (64 output cols + 16 halo each side), stride 98
  // so the 16 A-matrix rows land in 16 distinct LDS banks.
  __shared__ float l1 [16 * 98];
  __shared__ float l2 [16 * 98];
  __shared__ float l12[16 * 98];

  const int tid = threadIdx.x;
  const int bid = blockIdx.x;
  const int cg  = bid & 7;          // column group: 64 output cols
  const int rs  = (bid >> 3) & 31;  // row stripe: 16 rows
  const int p   = bid >> 8;         // plane (80)
  const int r0  = rs * 16;
  const int c0  = cg * 64;

  const float* i1 = img1 + (size_t)p * PLANE_ELEMS;
  const float* i2 = img2 + (size_t)p * PLANE_ELEMS;

  for (int idx = tid; idx < 16 * 96; idx += 128) {
    int lr = idx / 96, lc = idx - lr * 96;
    int gc = (c0 - 16 + lc) & (WW - 1);        // circular columns
    float a = i1[(r0 + lr) * WW + gc];
    float b = i2[(r0 + lr) * WW + gc];
    l1 [lr * 98 + lc] = a;
    l2 [lr * 98 + lc] = b;
    l12[lr * 98 + lc] = a * b;
  }
  __syncthreads();

  const int wave = tid >> 5;      // 4 waves, one 16x16 output tile each
  const int lane = tid & 31;
  const int m    = lane & 15;
  const int khi  = (lane >= 16) ? 2 : 0;  // A/B layout: upper lane-half holds K+2,K+3

  // Constant banded B fragments (4x16 f32 per K-chunk, 2 VGPRs).
  float bf[12][2];
  #pragma unroll
  for (int q = 0; q < 12; ++q) {
    bf[q][0] = wband(4 * q + 0 + khi, m);
    bf[q][1] = wband(4 * q + 1 + khi, m);
  }

  const int colbase = 16 * wave;
  float* srcs[3] = { l1, l2, l12 };
  float* dsts[3] = { h1, h2, h12 };

  #pragma unroll
  for (int f = 0; f < 3; ++f) {
    const float* L = srcs[f];
    v8f acc = {0.f, 0.f, 0.f, 0.f, 0.f, 0.f, 0.f, 0.f};
    #pragma unroll
    for (int q = 0; q < 12; ++q) {
      // A fragment: 16x4 f32 slice of [X_{t-1} | X_t | X_{t+1}]
      int base = m * 98 + colbase + 4 * q + khi;
      v2f a; a.x = L[base]; a.y = L[base + 1];
      v2f b; b.x = bf[q][0]; b.y = bf[q][1];
      acc = __builtin_amdgcn_wmma_f32_16x16x4_f32(
          false, a, false, b, (short)0, acc, false, false);
    }
    // C/D layout: VGPR v -> row (v + 8*laneHi), lane&15 -> col
    float* out = dsts[f] + (size_t)p * PLANE_ELEMS;
    int col   = c0 + colbase + m;
    int rbase = r0 + ((lane >= 16) ? 8 : 0);
    #pragma unroll
    for (int v = 0; v < 8; ++v)
      out[(rbase + v) * WW + col] = acc[v];
  }
}

__global__ __launch_bounds__(512)
void vpass_kernel(const float* __restrict__ h1,
                  const float* __restrict__ h2,
                  const float* __restrict__ h12,
                  double* __restrict__ partials)
{
  const int p = blockIdx.x;   // plane
  const int j = threadIdx.x;  // column (coalesced across threads)
  const float* a = h1  + (size_t)p * PLANE_ELEMS;
  const float* b = h2  + (size_t)p * PLANE_ELEMS;
  const float* c = h12 + (size_t)p * PLANE_ELEMS;

  // initial vertical window for output row 0: input rows -16..15 (mod 512)
  float s1 = 0.f, s2 = 0.f, s12 = 0.f;
  #pragma unroll 4
  for (int r = 0; r < 32; ++r) {
    int row = (r + 496) & 511;
    s1  += a[row * WW + j];
    s2  += b[row * WW + j];
    s12 += c[row * WW + j];
  }

  const float cw = (j == 0) ? 2.f : 1.f;  // col 512 duplicates col 0
  const float inv = 1.f / 1024.f;
  double acc = 0.0;

  for (int i = 0; i < HH; ++i) {
    float mu1 = s1 * inv, mu2 = s2 * inv, mu12 = s12 * inv;
    float mu1sq = mu1 * mu1, mu2sq = mu2 * mu2, mu1mu2 = mu1 * mu2;
    float sg1 = mu1 - mu1sq, sg2 = mu2 - mu2sq, sg12 = mu12 - mu1mu2;
    float s = sg1 + sg2, mm = mu1sq + mu2sq;
    float q = 1.f;
    if (s < EPSQ && mm > EPSQ) q = 2.f * mu1mu2 / mm;
    if (s > EPSQ && mm < EPSQ) q = 2.f * sg12 / s;
    if (s > EPSQ && mm > EPSQ) q = 4.f * mu1mu2 * sg12 / (mm * s);
    float w = cw * ((i == 0) ? 2.f : 1.f);  // row 512 duplicates row 0
    acc += (double)(q * w);

    int add = (i + 16)  & 511;
    int sub = (i + 496) & 511;  // i - 16 mod 512
    s1  += a[add * WW + j] - a[sub * WW + j];
    s2  += b[add * WW + j] - b[sub * WW + j];
    s12 += c[add * WW + j] - c[sub * WW + j];
  }

  __shared__ double red[512];
  red[j] = acc;
  __syncthreads();
  for (int t = 256; t > 0; t >>= 1) {
    if (j < t) red[j] += red[j + t];
    __syncthreads();
  }
  if (j == 0) partials[p] = red[0];
}

__global__ void final_kernel(const double* __restrict__ partials,
                             float* __restrict__ out)
{
  if (threadIdx.x == 0 && blockIdx.x == 0) {
    double t = 0.0;
    for (int i = 0; i < PLANES; ++i) t += partials[i];
    out[0] = (float)(t / (80.0 * 513.0 * 513.0));  // mean over [16,5,513,513]
  }
}

extern "C" void kernel_launch(void* const* d_in, const int* in_sizes, int n_in,
                              void* d_out, int out_size, void* d_ws, size_t ws_size,
                              hipStream_t stream)
{
  const float* img1 = (const float*)d_in[0];
  const float* img2 = (const float*)d_in[1];
  // d_in[2] is the constant 1/1024 box kernel; folded into the math.

  float* h1  = (float*)d_ws;
  float* h2  = h1 + (size_t)PLANES * PLANE_ELEMS;
  float* h12 = h2 + (size_t)PLANES * PLANE_ELEMS;
  double* partials = (double*)(h12 + (size_t)PLANES * PLANE_ELEMS);

  hpass_kernel<<<PLANES * 32 * 8, 128, 0, stream>>>(img1, img2, h1, h2, h12);
  vpass_kernel<<<PLANES, 512, 0, stream>>>(h1, h2, h12, partials);
  final_kernel<<<1, 32, 0, stream>>>(partials, (float*)d_out);
}